// GaussianRP_71811853189543
// MI455X (gfx1250) — compile-verified
//
#include <hip/hip_runtime.h>

// GaussianRP: x (8,16,32,1024) f32, log_sigma scalar ->
// rp (8,16,1024,1024) f32 = exp(-pairwise_sqdist / (2 sigma^2)) per (b,c).
//
// Gram matrix via V_WMMA_F32_16X16X4_F32 (K=32 as 8 chained k=4 steps).
// Store-bandwidth bound: 512 MB out @ 23.3 TB/s ~= 22 us floor.

typedef __attribute__((ext_vector_type(2))) float v2f;
typedef __attribute__((ext_vector_type(8))) float v8f;

#define TDIM 1024
#define DDIM 32
#define LOG2E 1.44269504088896340736f

__global__ __launch_bounds__(256) void gaussian_rp_wmma(
    const float* __restrict__ x, const float* __restrict__ log_sigma_p,
    float* __restrict__ out)
{
    const int bc   = blockIdx.y;          // 0..127  (b*16 + c)
    const int t0   = blockIdx.x * 16;     // row tile base
    const int lane = threadIdx.x & 31;
    const int wave = threadIdx.x >> 5;    // 0..7
    const int lo   = lane & 15;
    const int hi   = lane >> 4;           // 0 or 1

    const float* __restrict__ X = x + (size_t)bc * DDIM * TDIM;
    float* __restrict__ O = out + (size_t)bc * TDIM * TDIM;

    // scale = -log2(e) / (2 sigma^2) = -log2(e) * 0.5 * exp(-2*log_sigma)
    const float ls = log_sigma_p[0];
    const float scale = -LOG2E * 0.5f * __builtin_amdgcn_exp2f(-2.0f * ls * LOG2E);

    // ---- A fragments for row tile t0 (16x32 f32, eight 16x4 k-slices) ----
    // ISA layout 32-bit A 16x4: lane L -> m = L%16; VGPR0 = K=2*(L/16), VGPR1 = K+1.
    v2f a[8];
    float sqA = 0.0f;
#pragma unroll
    for (int kk = 0; kk < 8; ++kk) {
        const int d0 = 4 * kk + 2 * hi;
        const float ax = X[d0 * TDIM + t0 + lo];
        const float ay = X[(d0 + 1) * TDIM + t0 + lo];
        a[kk].x = ax;
        a[kk].y = ay;
        sqA += ax * ax + ay * ay;
    }
    // lane L and L^16 together cover all 32 d-values of column t0 + L%16
    sqA += __shfl_xor(sqA, 16, 32);

    // Row sqnorms needed in epilogue: C element (lane, vgpr v) is row m = v + 8*hi.
    float rowsq[8];
#pragma unroll
    for (int v = 0; v < 8; ++v)
        rowsq[v] = __shfl(sqA, v + 8 * hi, 32);

    // ---- sweep this wave's 8 column tiles (interleaved across waves) ----
#pragma unroll 1
    for (int i = 0; i < 8; ++i) {
        const int s0 = (i * 8 + wave) * 16;

        // B fragments (4x16 per k-slice): same per-lane indexing as A, base s0.
        v2f b[8];
        float sqB = 0.0f;
#pragma unroll
        for (int kk = 0; kk < 8; ++kk) {
            const int d0 = 4 * kk + 2 * hi;
            const float bx = X[d0 * TDIM + s0 + lo];
            const float by = X[(d0 + 1) * TDIM + s0 + lo];
            b[kk].x = bx;
            b[kk].y = by;
            sqB += bx * bx + by * by;
        }
        sqB += __shfl_xor(sqB, 16, 32);   // col sqnorm for n = lane%16

        // Gram tile: C = sum_k A_k * B_k  (8 x  v_wmma_f32_16x16x4_f32)
        v8f c = {};
#pragma unroll
        for (int kk = 0; kk < 8; ++kk)
            c = __builtin_amdgcn_wmma_f32_16x16x4_f32(
                    /*neg_a=*/false, a[kk], /*neg_b=*/false, b[kk],
                    /*c_mod=*/(short)0, c, /*reuse_a=*/false, /*reuse_b=*/false);

        // Epilogue: d2 = |t|^2 + |s|^2 - 2<t,s>, clamp, exp2(d2*scale), store.
#pragma unroll
        for (int v = 0; v < 8; ++v) {
            float d2 = rowsq[v] + sqB - 2.0f * c[v];
            d2 = fmaxf(d2, 0.0f);
            const float r = __builtin_amdgcn_exp2f(d2 * scale);
            O[(size_t)(t0 + v + 8 * hi) * TDIM + s0 + lo] = r;
        }
    }
}

extern "C" void kernel_launch(void* const* d_in, const int* in_sizes, int n_in,
                              void* d_out, int out_size, void* d_ws, size_t ws_size,
                              hipStream_t stream) {
    const float* x  = (const float*)d_in[0];
    const float* ls = (const float*)d_in[1];
    float* out = (float*)d_out;

    dim3 grid(TDIM / 16, 8 * 16);   // 64 row tiles x 128 (b,c) slabs
    dim3 block(256);                // 8 waves; wave w -> column tiles i*8+w
    gaussian_rp_wmma<<<grid, block, 0, stream>>>(x, ls, out);
}